// NovelModule_49512382988725
// MI455X (gfx1250) — compile-verified
//
#include <hip/hip_runtime.h>

// Problem constants (from reference)
#define NPROP 8192
#define DDIM  12544        // C*H*W = 256*7*7
#define FC    1024
#define KCLS  81
#define BGCLS 80
#define MPAD  96           // 81 padded to 6 WMMA tiles of 16
#define BIGF  3.0e38f

typedef __attribute__((ext_vector_type(2))) float v2f;
typedef __attribute__((ext_vector_type(8))) float v8f;

// ---------------------------------------------------------------------------
// Kernel 1: per-class iou weights + sums, bottom-10 background selection.
// Single workgroup of 1024 threads (32 waves). Uses LDS float atomics
// (ds_add_f32) and LDS tree reductions.
// ---------------------------------------------------------------------------
__global__ __launch_bounds__(1024) void k_prep(
    const float* __restrict__ ious, const int* __restrict__ gt,
    float* __restrict__ w, float* __restrict__ sum_w,
    int* __restrict__ present, int* __restrict__ bot_idx)
{
    __shared__ float s_iou[NPROP];      // masked iou (bg only, else +inf)
    __shared__ float s_sumw[MPAD];
    __shared__ float s_val[1024];
    __shared__ int   s_idx[1024];

    const int tid = threadIdx.x;
    if (tid < MPAD) s_sumw[tid] = 0.f;
    __syncthreads();

    for (int i = tid; i < NPROP; i += 1024) {
        float iou = ious[i];
        int   c   = gt[i];
        float wi  = (iou > 0.5f) ? iou : 0.f;
        w[i] = wi;
        if (wi > 0.f) atomicAdd(&s_sumw[c], wi);
        s_iou[i] = (c == BGCLS) ? iou : BIGF;
    }
    __syncthreads();

    // bottom-10 via repeated argmin
    for (int k = 0; k < 10; ++k) {
        float best = BIGF; int bi = 0;
        for (int i = tid; i < NPROP; i += 1024) {
            float v = s_iou[i];
            if (v < best) { best = v; bi = i; }
        }
        s_val[tid] = best; s_idx[tid] = bi;
        __syncthreads();
        for (int off = 512; off > 0; off >>= 1) {
            if (tid < off && s_val[tid + off] < s_val[tid]) {
                s_val[tid] = s_val[tid + off];
                s_idx[tid] = s_idx[tid + off];
            }
            __syncthreads();
        }
        if (tid == 0) {
            bot_idx[k] = s_idx[0];
            s_iou[s_idx[0]] = BIGF;   // exclude from next round
        }
        __syncthreads();
    }

    if (tid < KCLS) {
        float sw = s_sumw[tid];
        sum_w[tid]  = sw;
        present[tid] = (tid == BGCLS) ? 1 : (sw > 0.f ? 1 : 0);
    }
}

// ---------------------------------------------------------------------------
// Kernel 2: segment-sum of weighted features + bg mean + EMA fuse -> X.
// Each thread owns one feature index d and 81 per-class LDS accumulators.
// Wave-uniform skip of zero-weight proposals halves HBM traffic (~205 MB).
// X is [MPAD x DDIM], rows 81..95 zeroed (WMMA padding).
// ---------------------------------------------------------------------------
__global__ __launch_bounds__(128) void k_segsum(
    const float* __restrict__ bf, const float* __restrict__ proto,
    const int* __restrict__ gt, const float* __restrict__ w,
    const float* __restrict__ sum_w, const int* __restrict__ bot_idx,
    float* __restrict__ X)
{
    __shared__ float s_acc[128 * KCLS];   // 41.5 KB
    const int tid = threadIdx.x;
    const int d   = blockIdx.x * 128 + tid;

    float* acc = &s_acc[tid * KCLS];
    for (int c = 0; c < KCLS; ++c) acc[c] = 0.f;

    const float* col = bf + d;
    for (int i = 0; i < NPROP; ++i) {
        float wi = w[i];                  // uniform -> wave-coherent skip
        if (wi > 0.f) {
            int c = gt[i];
            acc[c] += wi * col[(size_t)i * DDIM];
        }
    }

    float bg = 0.f;
    for (int j = 0; j < 10; ++j)
        bg += col[(size_t)bot_idx[j] * DDIM];
    bg *= 0.1f;

    for (int c = 0; c < KCLS; ++c) {
        float p = proto[(size_t)c * DDIM + d];
        float val;
        if (c == BGCLS) {
            val = 0.5f * bg + 0.5f * p;
        } else {
            float sw = sum_w[c];
            val = (sw > 0.f)
                ? 0.5f * (acc[c] / fmaxf(sw, 1e-12f)) + 0.5f * p
                : p;
        }
        X[(size_t)c * DDIM + d] = val;
    }
    for (int c = KCLS; c < MPAD; ++c) X[(size_t)c * DDIM + d] = 0.f;
}

// ---------------------------------------------------------------------------
// Kernels 3/4: f32 WMMA GEMM, one 16x16 output tile per wave.
//   C[MPAD x NN] = A[MPAD x K] @ B[K x NN]   (row-major)
// K and NN are template constants so the inner loop strength-reduces to
// pointer bumps + immediate-offset global loads (no per-iter 64-bit muls).
// Fragment layout per CDNA5 ISA 7.12.2:
//   A 16x4 f32 : lane l(0..15)=M, half-wave selects K{0,1} vs K{2,3}
//   B 4x16 f32 : VGPR r holds row K=r (lanes 0..15) / K=r+2 (lanes 16..31)
//   C/D 16x16  : VGPR r holds row M=r (lanes 0..15) / M=r+8 (lanes 16..31)
// MODE 0: out = relu(acc + bias)                          -> hidden buffer
// MODE 1: out = present ? 0.5*relu(acc+bias)+0.5*stored
//                       : stored        (rows < 81 only)  -> d_out
// ---------------------------------------------------------------------------
template<int K, int NN, int MODE>
__global__ __launch_bounds__(128) void k_gemm_wmma(
    const float* __restrict__ A, const float* __restrict__ B,
    const float* __restrict__ bias, float* __restrict__ C,
    const int* __restrict__ present, const float* __restrict__ stored)
{
    constexpr int NMT = MPAD / 16;          // 6 m-tiles
    const int wave = blockIdx.x * (blockDim.x >> 5) + (threadIdx.x >> 5);
    const int lane = threadIdx.x & 31;
    const int half = lane >> 4;             // 0: lanes 0-15, 1: lanes 16-31
    const int l    = lane & 15;

    const int mt = wave % NMT;
    const int nt = wave / NMT;
    const int m0 = mt * 16;
    const int n0 = nt * 16;
    const int koff = 2 * half;

    // per-lane base pointers (compile-time strides)
    const float* Ap = A + (m0 + l) * K + koff;       // +k : consecutive 8B
    const float* Bp = B + koff * NN + (n0 + l);      // +k*NN : column walk

    v8f acc = {};
    #pragma unroll 8
    for (int k = 0; k < K; k += 4) {
        v2f a = *reinterpret_cast<const v2f*>(Ap + k);
        v2f b;
        b.x = Bp[k * NN];
        b.y = Bp[k * NN + NN];
        acc = __builtin_amdgcn_wmma_f32_16x16x4_f32(
                  false, a, false, b, (short)0, acc, false, false);
    }

    const float bn = bias[n0 + l];
    #pragma unroll
    for (int r = 0; r < 8; ++r) {
        int   m = m0 + r + 8 * half;
        int   n = n0 + l;
        float v = acc[r] + bn;
        v = v > 0.f ? v : 0.f;              // relu
        if (MODE == 0) {
            C[m * NN + n] = v;
        } else if (m < KCLS) {
            float st = stored[m * NN + n];
            C[m * NN + n] = present[m] ? (0.5f * v + 0.5f * st) : st;
        }
    }
}

// ---------------------------------------------------------------------------
extern "C" void kernel_launch(void* const* d_in, const int* in_sizes, int n_in,
                              void* d_out, int out_size, void* d_ws, size_t ws_size,
                              hipStream_t stream) {
    const float* box_features = (const float*)d_in[0];   // [8192,256,7,7]
    const int*   gt_classes   = (const int*)  d_in[1];   // [8192]
    const float* ious         = (const float*)d_in[2];   // [8192]
    const float* prototypes   = (const float*)d_in[3];   // [81,256,7,7]
    const float* proto_feats  = (const float*)d_in[4];   // [81,1024]
    const float* W1           = (const float*)d_in[5];   // [12544,1024]
    const float* b1           = (const float*)d_in[6];   // [1024]
    const float* W2           = (const float*)d_in[7];   // [1024,1024]
    const float* b2           = (const float*)d_in[8];   // [1024]
    float* out = (float*)d_out;                          // [81,1024]

    // workspace layout
    char*  ws      = (char*)d_ws;
    float* w       = (float*)(ws);                         // 8192 f32
    float* sum_w   = (float*)(ws + 32768);                 // 96 f32
    int*   present = (int*)  (ws + 33152);                 // 96 i32
    int*   bot_idx = (int*)  (ws + 33536);                 // 16 i32
    float* X       = (float*)(ws + 33792);                 // 96*12544 f32
    float* H       = (float*)(ws + 33792 + (size_t)MPAD * DDIM * 4); // 96*1024 f32

    // 1) weights, per-class sums, bottom-10 bg
    k_prep<<<1, 1024, 0, stream>>>(ious, gt_classes, w, sum_w, present, bot_idx);

    // 2) segment sum + bg mean + EMA fuse -> X [96 x 12544]
    k_segsum<<<DDIM / 128, 128, 0, stream>>>(
        box_features, prototypes, gt_classes, w, sum_w, bot_idx, X);

    // 3) H = relu(X @ W1 + b1)   (96x12544 @ 12544x1024)
    //    6 m-tiles * 64 n-tiles = 384 waves = 96 blocks of 4 waves
    k_gemm_wmma<DDIM, FC, 0><<<96, 128, 0, stream>>>(
        X, W1, b1, H, nullptr, nullptr);

    // 4) out = EMA(present, relu(H @ W2 + b2), proto_feats)   rows < 81
    k_gemm_wmma<FC, FC, 1><<<96, 128, 0, stream>>>(
        H, W2, b2, out, present, proto_feats);
}